// AxonalConnections_25907242729986
// MI455X (gfx1250) — compile-verified
//
#include <hip/hip_runtime.h>

typedef float v2f __attribute__((ext_vector_type(2)));
typedef float v8f __attribute__((ext_vector_type(8)));

// Problem geometry (fixed by the reference)
constexpr int kB   = 64;     // batch
constexpr int kH   = 512;    // image height/width
constexpr int kW   = 512;
constexpr int kHP  = 64;     // patches per side
constexpr int kNP  = kHP * kHP;  // 4096 patches
constexpr int kS   = 64;     // source patch size (8x8)
constexpr int kT   = 64;     // target patch size (8x8)
constexpr int kPad = 68;     // padded LDS row stride (floats) -> conflict-free fragment reads

// One workgroup per patch: 256 threads = 8 wave32.
// Computes X = A (64x64, batch x pixels) * W^T (pixels x targets) with
// V_WMMA_F32_16X16X4_F32 tiles, then gate/bias/mask epilogue + folded store.
__global__ __launch_bounds__(256) void axonal_wmma_f32_kernel(
    const float* __restrict__ src,        // [B, 512, 512]
    const float* __restrict__ transforms, // [NP, 64, 64]
    const float* __restrict__ gates,      // [NP]
    const float* __restrict__ biases,     // [NP]
    float* __restrict__ out)              // [B, 512, 512]
{
    __shared__ float Alds[kB * kPad];   // A_p : [b][s], padded rows
    __shared__ float Wlds[kT * kPad];   // W_p : [t][s], padded rows
    __shared__ float maskLds[kB];

    const int p   = blockIdx.x;   // patch id
    const int pi  = p / kHP;
    const int pj  = p % kHP;
    const int tid = threadIdx.x;  // 0..255

    // ---- Stage A_p: gather 64 (b) x 64 (s) from the 8x8 patch of each image ----
    // 1024 float4 chunks, 4 per thread; each chunk = 4 contiguous pixels of one patch row.
    const float* srcPatch = src + (size_t)pi * 8 * kW + (size_t)pj * 8;
#pragma unroll
    for (int it = 0; it < 4; ++it) {
        const int v  = tid + it * 256;      // 0..1023
        const int b  = v >> 4;              // image index
        const int r  = (v & 15) >> 1;       // patch row 0..7
        const int c4 = (v & 1) * 4;         // 0 or 4
        const float4 d = *(const float4*)(srcPatch + (size_t)b * kH * kW
                                                   + (size_t)r * kW + c4);
        float* dst = &Alds[b * kPad + r * 8 + c4];
        dst[0] = d.x; dst[1] = d.y; dst[2] = d.z; dst[3] = d.w;
    }

    // ---- Stage W_p: 64x64 contiguous, fully coalesced float4 loads ----
    const float* wp = transforms + (size_t)p * (kT * kS);
#pragma unroll
    for (int it = 0; it < 4; ++it) {
        const int v  = tid + it * 256;      // 0..1023
        const int t  = v >> 4;
        const int s4 = (v & 15) * 4;
        const float4 d = *(const float4*)(wp + (size_t)v * 4);
        float* dst = &Wlds[t * kPad + s4];
        dst[0] = d.x; dst[1] = d.y; dst[2] = d.z; dst[3] = d.w;
    }
    __syncthreads();

    // ---- activity mask per image: strength = sum_s A[b][s] ----
    if (tid < kB) {
        const float* row = &Alds[tid * kPad];
        float sum = 0.f;
#pragma unroll
        for (int s = 0; s < kS; ++s) sum += row[s];
        maskLds[tid] = (sum > 0.f) ? 1.f : 0.f;
    }
    __syncthreads();

    const float gate = gates[p];
    const float bias = biases[p];

    const int wave = tid >> 5;   // 0..7
    const int lane = tid & 31;   // wave32 lane
    const int lr   = lane & 15;  // 0..15
    const int lh   = lane >> 4;  // 0 or 1

    float* outPatch = out + (size_t)pi * 8 * kW + (size_t)pj * 8;

#pragma unroll
    for (int i = 0; i < 2; ++i) {
        const int tile = wave * 2 + i;  // 0..15 -> 4x4 grid of 16x16 output tiles
        const int bm = tile >> 2;       // block row (batch dim)
        const int tn = tile & 3;        // block col (target dim)

        v8f acc = {};
        // A fragment 16x4 (M=lane%16, K = 2*(lane/16) + vgpr) -> contiguous float2
        const float* arow = &Alds[(bm * 16 + lr) * kPad + 2 * lh];
        // B fragment 4x16: B[k][n] = W[t=n][s=k] (N=lane%16, K = 2*(lane/16) + vgpr)
        const float* brow = &Wlds[(tn * 16 + lr) * kPad + 2 * lh];

#pragma unroll
        for (int k0 = 0; k0 < kS; k0 += 4) {
            v2f a = *(const v2f*)(arow + k0);
            v2f b = *(const v2f*)(brow + k0);
            // 8 args: (neg_a, A, neg_b, B, c_mod, C, reuse_a, reuse_b)
            acc = __builtin_amdgcn_wmma_f32_16x16x4_f32(
                false, a, false, b, (short)0, acc, false, false);
        }

        // ---- epilogue: (x*gate + bias) * mask, folded scatter ----
        const int tcol = tn * 16 + lr;       // target pixel index 0..63
        const int tr = tcol >> 3, tc = tcol & 7;
#pragma unroll
        for (int e = 0; e < 8; ++e) {
            const int bidx = bm * 16 + lh * 8 + e;  // C/D layout: row = vgpr + 8*(lane/16)
            const float val = (acc[e] * gate + bias) * maskLds[bidx];
            outPatch[(size_t)bidx * kH * kW + (size_t)tr * kW + tc] = val;
        }
    }
}

extern "C" void kernel_launch(void* const* d_in, const int* in_sizes, int n_in,
                              void* d_out, int out_size, void* d_ws, size_t ws_size,
                              hipStream_t stream) {
    const float* src        = (const float*)d_in[0];
    const float* transforms = (const float*)d_in[1];
    const float* gates      = (const float*)d_in[2];
    const float* biases     = (const float*)d_in[3];
    float* out              = (float*)d_out;

    axonal_wmma_f32_kernel<<<dim3(kNP), dim3(256), 0, stream>>>(
        src, transforms, gates, biases, out);
}